// HarMABase_46840913330248
// MI455X (gfx1250) — compile-verified
//
#include <hip/hip_runtime.h>

typedef float v2f __attribute__((ext_vector_type(2)));
typedef float v8f __attribute__((ext_vector_type(8)));
typedef unsigned int u32x4 __attribute__((ext_vector_type(4)));
typedef int i32x4 __attribute__((ext_vector_type(4)));
typedef int i32x8 __attribute__((ext_vector_type(8)));

#define B_N  4096
#define D_K  512
#define NCLS 64

// workspace layout (float offsets)
#define WS_ROWSUM   0                        // [B]  sum_j exp(logits[i,j])
#define WS_COLSUM   (WS_ROWSUM + B_N)        // [B]  sum_i exp(logits[i,j])
#define WS_DIAG     (WS_COLSUM + B_N)        // [B]  logits[i,i]
#define WS_ROWSUMW  (WS_DIAG + B_N)          // [B]  sum_c cnt_c exp(s[i,c])
#define WS_DIAG_S   (WS_ROWSUMW + B_N)       // [B]  s[i,label[i]]
#define WS_DIAG_T   (WS_DIAG_S + B_N)        // [B]  t[i,label[i]]
#define WS_COLSEXP  (WS_DIAG_T + B_N)        // [64] sum_i exp(t[i,c])
#define WS_CNT      (WS_COLSEXP + NCLS)      // [64] class counts (float)
#define WS_TXTC     (WS_CNT + NCLS)          // [64*512] text centroids
#define WS_IMGC     (WS_TXTC + NCLS*D_K)     // [64*512] image centroids
#define WS_ZEROCNT  WS_CNT                   // floats to zero each call

#define WMMA_F32(a, b, c) \
    __builtin_amdgcn_wmma_f32_16x16x4_f32(false, (a), false, (b), (short)0, (c), false, false)

// TDM: toolchains differ in arity (ROCm7.2/clang-22: 5 args; clang-23: 6 args)
#if __clang_major__ >= 23
#define TDM_LOAD(g0, g1, z4, z8) __builtin_amdgcn_tensor_load_to_lds((g0), (g1), (z4), (z4), (z8), 0)
#else
#define TDM_LOAD(g0, g1, z4, z8) __builtin_amdgcn_tensor_load_to_lds((g0), (g1), (z4), (z4), 0)
#endif

__global__ void zero_kernel(float* p, int n) {
    int i = blockIdx.x * blockDim.x + threadIdx.x;
    if (i < n) p[i] = 0.0f;
}

// One block per class: gather class sums and counts, write centroids.
__global__ void class_centers_kernel(const float* __restrict__ img,
                                     const float* __restrict__ txt,
                                     const int* __restrict__ label,
                                     float* __restrict__ ws) {
    __shared__ int lbl[B_N];
    const int c = blockIdx.x;
    for (int j = threadIdx.x; j < B_N; j += blockDim.x) lbl[j] = label[j];
    __syncthreads();

    const int d = threadIdx.x;           // handles dims d and d+256
    float si0 = 0.f, si1 = 0.f, st0 = 0.f, st1 = 0.f;
    int cnt = 0;
    for (int j = 0; j < B_N; ++j) {
        if (lbl[j] == c) {
            ++cnt;
            si0 += img[j * D_K + d];       si1 += img[j * D_K + d + 256];
            st0 += txt[j * D_K + d];       st1 += txt[j * D_K + d + 256];
        }
    }
    const float inv = cnt > 0 ? 1.0f / (float)cnt : 0.0f;  // empty class -> zero centroid
    float* imgc = ws + WS_IMGC;
    float* txtc = ws + WS_TXTC;
    imgc[c * D_K + d] = si0 * inv;  imgc[c * D_K + d + 256] = si1 * inv;
    txtc[c * D_K + d] = st0 * inv;  txtc[c * D_K + d + 256] = st1 * inv;
    if (threadIdx.x == 0) ws[WS_CNT + c] = (float)cnt;
}

// logits diagonal: one wave per row
__global__ void diag_kernel(const float* __restrict__ img, const float* __restrict__ txt,
                            const float* __restrict__ temp, float* __restrict__ ws) {
    const int lane = threadIdx.x & 31;
    const int row  = blockIdx.x * 8 + (threadIdx.x >> 5);
    float s = 0.f;
    for (int k = lane; k < D_K; k += 32) s += img[row * D_K + k] * txt[row * D_K + k];
    for (int m = 16; m; m >>= 1) s += __shfl_xor(s, m, 32);
    if (lane == 0) ws[WS_DIAG + row] = s / temp[0];
}

// epilogue for one 16x16 accumulator: exp, row/col sum-of-exp reductions
__device__ __forceinline__ void tile_epilogue(const v8f& acc, float invT,
                                              int row0, int col0,
                                              int mr, int hi, int lane,
                                              float* rowsum, float* colsum) {
    float csum = 0.f;
#pragma unroll
    for (int r = 0; r < 8; ++r) {
        float e = __expf(acc[r] * invT);   // element (row0 + r + 8*hi, col0 + mr)
        csum += e;
        float rs = e;
        rs += __shfl_xor(rs, 1, 32);
        rs += __shfl_xor(rs, 2, 32);
        rs += __shfl_xor(rs, 4, 32);
        rs += __shfl_xor(rs, 8, 32);
        if (mr == 0) atomicAdd(&rowsum[row0 + r + hi * 8], rs);
    }
    csum += __shfl_xor(csum, 16, 32);
    if (lane < 16) atomicAdd(&colsum[col0 + lane], csum);
}

// ---------------- main contrastive GEMM ----------------
// Block tile: 64 rows x 128 cols. K chunked by 64. Panels staged in LDS by the
// Tensor Data Mover (pad mode -> LDS row stride 66 floats, bank-conflict free).
// Each wave computes a 32x32 tile = 4 f32 WMMA accumulators (register reuse 2x2).
#define BLK_R      64
#define BLK_C      128
#define KC         64
#define LDS_STRIDE 66
// group1 dword0: data_size=4B (2<<16) | pad_enable (1<<20) | pad_interval=64dw (5<<22)
//                | pad_amount=2dw (1<<25)
#define G1DW0 ((2 << 16) | (1 << 20) | (5 << 22) | (1 << 25))

__global__ void __launch_bounds__(256)
contr_tiles_kernel(const float* __restrict__ img,
                   const float* __restrict__ txt,
                   const float* __restrict__ temp,
                   float* __restrict__ ws) {
    __shared__ float ldsA[BLK_R * LDS_STRIDE];
    __shared__ float ldsB[BLK_C * LDS_STRIDE];

    const int lane = threadIdx.x & 31;
    const int wave = threadIdx.x >> 5;
    const int mr = lane & 15, hi = lane >> 4;
    const int bR0 = (int)(blockIdx.x >> 5) * BLK_R;   // 64 row blocks
    const int bC0 = (int)(blockIdx.x & 31) * BLK_C;   // 32 col blocks
    const int wr = wave >> 2, wc = wave & 3;          // 2x4 waves -> 32x32 each

    const float* Abase = ldsA + (wr * 32 + mr) * LDS_STRIDE + 2 * hi;
    const float* Bbase = ldsB + (wc * 32 + mr) * LDS_STRIDE + 2 * hi;

    // LDS byte offsets for TDM descriptors (low 32 bits of generic LDS address)
    const unsigned ldsA_off = (unsigned)(unsigned long long)(void*)ldsA;
    const unsigned ldsB_off = (unsigned)(unsigned long long)(void*)ldsB;

    v8f acc00 = {}, acc01 = {}, acc10 = {}, acc11 = {};
    const i32x4 z4 = {};
    const i32x8 z8 = {};
    (void)z8;

    for (int ch = 0; ch < D_K / KC; ++ch) {
        __syncthreads();                 // previous chunk fully consumed
        if (wave == 0) {
            unsigned long long gaA = (unsigned long long)img +
                                     ((unsigned long long)(bR0 * D_K + ch * KC) << 2);
            unsigned long long gaB = (unsigned long long)txt +
                                     ((unsigned long long)(bC0 * D_K + ch * KC) << 2);
            // D# group0: count=1 | lds_addr | global_addr | type=2 in bits 127:126
            u32x4 g0A = { 1u, ldsA_off, (unsigned)gaA,
                          (unsigned)((gaA >> 32) & 0x01ffffffu) | 0x80000000u };
            u32x4 g0B = { 1u, ldsB_off, (unsigned)gaB,
                          (unsigned)((gaB >> 32) & 0x01ffffffu) | 0x80000000u };
            // D# group1: dims: tensor 512 x 4096 (row-major, stride 512), tile KC x rows
            i32x8 g1A = { G1DW0, (int)(D_K << 16), (int)(B_N << 16), (int)(KC << 16),
                          BLK_R, D_K, 0, 0 };
            i32x8 g1B = { G1DW0, (int)(D_K << 16), (int)(B_N << 16), (int)(KC << 16),
                          BLK_C, D_K, 0, 0 };
            TDM_LOAD(g0A, g1A, z4, z8);
            TDM_LOAD(g0B, g1B, z4, z8);
            __builtin_amdgcn_s_wait_tensorcnt(0);
        }
        __syncthreads();                 // panels visible to all waves

#pragma unroll 4
        for (int kk = 0; kk < KC; kk += 4) {
            v2f a0 = *(const v2f*)(Abase + kk);
            v2f a1 = *(const v2f*)(Abase + 16 * LDS_STRIDE + kk);
            v2f b0 = *(const v2f*)(Bbase + kk);
            v2f b1 = *(const v2f*)(Bbase + 16 * LDS_STRIDE + kk);
            acc00 = WMMA_F32(a0, b0, acc00);
            acc01 = WMMA_F32(a0, b1, acc01);
            acc10 = WMMA_F32(a1, b0, acc10);
            acc11 = WMMA_F32(a1, b1, acc11);
        }
    }

    const float invT = 1.0f / temp[0];
    float* rowsum = ws + WS_ROWSUM;
    float* colsum = ws + WS_COLSUM;
    const int r0 = bR0 + wr * 32, c0 = bC0 + wc * 32;
    tile_epilogue(acc00, invT, r0,      c0,      mr, hi, lane, rowsum, colsum);
    tile_epilogue(acc01, invT, r0,      c0 + 16, mr, hi, lane, rowsum, colsum);
    tile_epilogue(acc10, invT, r0 + 16, c0,      mr, hi, lane, rowsum, colsum);
    tile_epilogue(acc11, invT, r0 + 16, c0 + 16, mr, hi, lane, rowsum, colsum);
}

// ---------------- affiliation GEMMs ----------------
// Each wave computes 16 rows x all 64 classes (4 accumulators, A fragment reused 4x).
// Waves 0..255: s = image @ txt_centers^T ; waves 256..511: t = text @ img_centers^T
__global__ void __launch_bounds__(256)
affil_tiles_kernel(const float* __restrict__ img,
                   const float* __restrict__ txt,
                   const int* __restrict__ label,
                   const float* __restrict__ temp2,
                   float* __restrict__ ws) {
    const int lane = threadIdx.x & 31;
    const int wave = threadIdx.x >> 5;
    const int t = blockIdx.x * 8 + wave;       // 0..511 (wave-uniform)
    const int isT = t >> 8;
    const int row0 = (t & 255) << 4;
    const int mr = lane & 15, hi = lane >> 4;

    const float* A  = isT ? txt : img;
    const float* Bc = ws + (isT ? WS_IMGC : WS_TXTC);
    const float* arow = A  + (size_t)(row0 + mr) * D_K + 2 * hi;
    const float* brow = Bc + (size_t)mr * D_K + 2 * hi;

    v8f acc[4] = { {}, {}, {}, {} };
    for (int k = 0; k < D_K; k += 4) {
        v2f a = *(const v2f*)(arow + k);
#pragma unroll
        for (int cg = 0; cg < 4; ++cg) {
            v2f b = *(const v2f*)(brow + cg * 16 * D_K + k);
            acc[cg] = WMMA_F32(a, b, acc[cg]);
        }
    }

    const float invT = 1.0f / temp2[0];
    const float* cntf = ws + WS_CNT;
    int lab[8];
#pragma unroll
    for (int r = 0; r < 8; ++r) lab[r] = label[row0 + r + hi * 8];

    if (!isT) {
        float wrow[8] = { 0.f, 0.f, 0.f, 0.f, 0.f, 0.f, 0.f, 0.f };
#pragma unroll
        for (int cg = 0; cg < 4; ++cg) {
            const int col = cg * 16 + mr;
            const float cw = cntf[col];
#pragma unroll
            for (int r = 0; r < 8; ++r) {
                float val = acc[cg][r] * invT;
                if (lab[r] == col) ws[WS_DIAG_S + row0 + r + hi * 8] = val;
                wrow[r] += cw * __expf(val);
            }
        }
#pragma unroll
        for (int r = 0; r < 8; ++r) {
            float w = wrow[r];
            w += __shfl_xor(w, 1, 32);
            w += __shfl_xor(w, 2, 32);
            w += __shfl_xor(w, 4, 32);
            w += __shfl_xor(w, 8, 32);
            if (mr == 0) atomicAdd(ws + WS_ROWSUMW + row0 + r + hi * 8, w);
        }
    } else {
#pragma unroll
        for (int cg = 0; cg < 4; ++cg) {
            const int col = cg * 16 + mr;
            float csum = 0.f;
#pragma unroll
            for (int r = 0; r < 8; ++r) {
                float val = acc[cg][r] * invT;
                if (lab[r] == col) ws[WS_DIAG_T + row0 + r + hi * 8] = val;
                csum += __expf(val);
            }
            csum += __shfl_xor(csum, 16, 32);
            if (lane < 16) atomicAdd(ws + WS_COLSEXP + cg * 16 + lane, csum);
        }
    }
}

__global__ void finalize_kernel(const int* __restrict__ label,
                                const float* __restrict__ ws,
                                float* __restrict__ out) {
    __shared__ float red[256];
    float s = 0.f;
    for (int i = threadIdx.x; i < B_N; i += blockDim.x) {
        s += ws[WS_DIAG + i]   - logf(ws[WS_ROWSUM + i]);            // i2t
        s += ws[WS_DIAG + i]   - logf(ws[WS_COLSUM + i]);            // t2i
        s += ws[WS_DIAG_S + i] - logf(ws[WS_ROWSUMW + i]);           // affil i2t
        s += ws[WS_DIAG_T + i] - logf(ws[WS_COLSEXP + label[i]]);    // affil t2i
    }
    red[threadIdx.x] = s;
    __syncthreads();
    for (int off = 128; off; off >>= 1) {
        if ((int)threadIdx.x < off) red[threadIdx.x] += red[threadIdx.x + off];
        __syncthreads();
    }
    if (threadIdx.x == 0) out[0] = -red[0] / (2.0f * (float)B_N);
}

extern "C" void kernel_launch(void* const* d_in, const int* in_sizes, int n_in,
                              void* d_out, int out_size, void* d_ws, size_t ws_size,
                              hipStream_t stream) {
    const float* img   = (const float*)d_in[0];
    const float* txt   = (const float*)d_in[1];
    const int*   label = (const int*)d_in[2];
    const float* temp  = (const float*)d_in[3];
    const float* temp2 = (const float*)d_in[4];
    float* ws  = (float*)d_ws;
    float* out = (float*)d_out;

    zero_kernel<<<(WS_ZEROCNT + 255) / 256, 256, 0, stream>>>(ws, WS_ZEROCNT);
    class_centers_kernel<<<NCLS, 256, 0, stream>>>(img, txt, label, ws);
    diag_kernel<<<B_N / 8, 256, 0, stream>>>(img, txt, temp, ws);
    contr_tiles_kernel<<<(B_N / BLK_R) * (B_N / BLK_C), 256, 0, stream>>>(img, txt, temp, ws);
    affil_tiles_kernel<<<64, 256, 0, stream>>>(img, txt, label, temp2, ws);
    finalize_kernel<<<1, 256, 0, stream>>>(label, ws, out);
}